// UpSample_79310866088302
// MI455X (gfx1250) — compile-verified
//
#include <hip/hip_runtime.h>
#include <hip/hip_bf16.h>
#include <math.h>

typedef float v2f __attribute__((ext_vector_type(2)));
typedef float v8f __attribute__((ext_vector_type(8)));

// ---------------------------------------------------------------------------
// V_WMMA_F32_16X16X4_F32 wrapper (wave32).
// A (16x4): lane l: m = l&15, half = l>>4, a[v] = A[m][2*half+v]
// B (4x16): b[v] = B[2*half+v][l&15]
// C/D (16x16): c[r] = C[r + 8*half][l&15]
// ---------------------------------------------------------------------------
__device__ inline v8f wmma4(v2f a, v2f b, v8f c) {
    return __builtin_amdgcn_wmma_f32_16x16x4_f32(
        /*neg_a=*/false, a, /*neg_b=*/false, b,
        /*c_mod=*/(short)0, c, /*reuse_a=*/false, /*reuse_b=*/false);
}

__device__ inline float gelu_tanh(float x) {
    float x3 = x * x * x;
    return 0.5f * x * (1.0f + tanhf(0.7978845608028654f * (x + 0.044715f * x3)));
}

#define BS 16
#define N_STK 256
#define N_BEF 512
#define C_COOR 128
#define SP_IN 512
#define SP_OUT 256
#define DN_CH 128
#define N_PNT 32
#define LSEQ 16384           // n_bef * n_pnt
#define LDS_STRIDE 72        // (2*72)%64 == 16 -> half-wave bank groups disjoint

// ---------------------------------------------------------------------------
// Kernel 1: pairwise sq-distance + top-2 + inverse-distance weights.
// One wave per (b, bef-row). 1024 blocks x 256 threads (8 waves).
// ---------------------------------------------------------------------------
__global__ __launch_bounds__(256) void topk_kernel(
    const float* __restrict__ stk_coor, const float* __restrict__ stk_coor_bef,
    int* __restrict__ idx0, int* __restrict__ idx1,
    float* __restrict__ wA, float* __restrict__ wB)
{
    __shared__ float befs[8][C_COOR];
    const int wv = threadIdx.x >> 5, lane = threadIdx.x & 31;
    const int row = blockIdx.x * 8 + wv;           // 0..8191
    const int b = row >> 9, r = row & 511;

    const float* bef = stk_coor_bef + ((size_t)b * N_BEF + r) * C_COOR;
    #pragma unroll
    for (int i = 0; i < 4; ++i) befs[wv][lane * 4 + i] = bef[lane * 4 + i];
    __syncthreads();

    float bef2 = 0.f;
    #pragma unroll 8
    for (int k = 0; k < C_COOR; ++k) bef2 = fmaf(befs[wv][k], befs[wv][k], bef2);

    float d1 = 3.4e38f, d2 = 3.4e38f;
    int i1 = 0, i2 = 0;
    for (int jj = 0; jj < 8; ++jj) {
        const int j = jj * 32 + lane;
        const float* st = stk_coor + ((size_t)b * N_STK + j) * C_COOR;
        float dot = 0.f, s2 = 0.f;
        #pragma unroll 8
        for (int k = 0; k < C_COOR; ++k) {
            const float s = st[k];
            dot = fmaf(befs[wv][k], s, dot);
            s2  = fmaf(s, s, s2);
        }
        const float d = bef2 + s2 - 2.f * dot;
        if (d < d1)      { d2 = d1; i2 = i1; d1 = d; i1 = j; }
        else if (d < d2) { d2 = d;  i2 = j; }
    }
    // wave32 merge of sorted (d1,d2) pairs
    #pragma unroll
    for (int off = 16; off >= 1; off >>= 1) {
        float od1 = __shfl_down(d1, off, 32);
        int   oi1 = __shfl_down(i1, off, 32);
        float od2 = __shfl_down(d2, off, 32);
        int   oi2 = __shfl_down(i2, off, 32);
        if (od1 < d1) { float td = d1; int ti = i1; d1 = od1; i1 = oi1; od1 = td; oi1 = ti; }
        if (od1 < d2) { d2 = od1; i2 = oi1; }
        if (od2 < d2) { d2 = od2; i2 = oi2; }
    }
    if (lane == 0) {
        const float r1 = 1.0f / (d1 + 1e-8f);
        const float r2 = 1.0f / (d2 + 1e-8f);
        const float inv = 1.0f / (r1 + r2);
        idx0[row] = i1; idx1[row] = i2;
        wA[row] = r1 * inv; wB[row] = r2 * inv;
    }
}

// ---------------------------------------------------------------------------
// Kernel 2: repack ConvTranspose weights into phase GEMM matrices.
// Wcat[par][o][k2], k2 in [0,256): even phase taps {3,1}, odd phase taps {2,0}.
// ---------------------------------------------------------------------------
__global__ __launch_bounds__(256) void wcat_kernel(
    const float* __restrict__ w_ct, float* __restrict__ wcat)
{
    const int t = blockIdx.x * 256 + threadIdx.x;   // 0..65535
    const int par = t >> 15;
    const int o   = (t >> 8) & 127;
    const int k2  = t & 255;
    const int c   = k2 & 127;
    const int tap = (par == 0) ? (k2 < 128 ? 3 : 1) : (k2 < 128 ? 2 : 0);
    wcat[t] = w_ct[((size_t)c * DN_CH + o) * 4 + tap];
}

// ---------------------------------------------------------------------------
// Kernel 3: sparse branch. Per block: C[256 x 64] = W_sp[256x512] * X[512x64],
// X = gathered+interpolated sparse features (fused into LDS staging).
// 8 waves: wave w owns M rows [32w,32w+32) as 2 x (16x16) x 4 N-subtiles.
// ---------------------------------------------------------------------------
__global__ __launch_bounds__(256) void sparse_gemm_kernel(
    const float* __restrict__ sparse_fea, const float* __restrict__ w_sp,
    const float* __restrict__ b_sp, const float* __restrict__ gamma_sp,
    const float* __restrict__ beta_sp,
    const int* __restrict__ idx0, const int* __restrict__ idx1,
    const float* __restrict__ wA, const float* __restrict__ wB,
    float* __restrict__ out)
{
    __shared__ float Xs[32 * LDS_STRIDE];
    const int b = blockIdx.y;
    const int n0 = blockIdx.x * 64;
    const int tid = threadIdx.x;
    const int wv = tid >> 5, lane = tid & 31, half = lane >> 4, l15 = lane & 15;
    const int tn = tid & 63, trow = tid >> 6;
    const int ng = n0 + tn;

    const int   gi0 = idx0[b * N_BEF + ng];
    const int   gi1 = idx1[b * N_BEF + ng];
    const float gw0 = wA[b * N_BEF + ng];
    const float gw1 = wB[b * N_BEF + ng];
    const float* feab = sparse_fea + (size_t)b * SP_IN * N_STK;

    v8f acc[2][4];
    #pragma unroll
    for (int i = 0; i < 2; ++i)
        #pragma unroll
        for (int j = 0; j < 4; ++j) { v8f z = {}; acc[i][j] = z; }

    const int m0 = wv * 32;
    for (int kk = 0; kk < SP_IN; kk += 32) {
        __syncthreads();
        #pragma unroll
        for (int i = 0; i < 8; ++i) {
            const int cr = trow + i * 4;
            const float* fc = feab + (size_t)(kk + cr) * N_STK;
            Xs[cr * LDS_STRIDE + tn] = gw0 * fc[gi0] + gw1 * fc[gi1];
        }
        __syncthreads();
        #pragma unroll
        for (int k4 = 0; k4 < 8; ++k4) {
            const int kg = kk + k4 * 4 + 2 * half;
            v2f a0, a1;
            a0.x = w_sp[(size_t)(m0 + l15) * SP_IN + kg];
            a0.y = w_sp[(size_t)(m0 + l15) * SP_IN + kg + 1];
            a1.x = w_sp[(size_t)(m0 + 16 + l15) * SP_IN + kg];
            a1.y = w_sp[(size_t)(m0 + 16 + l15) * SP_IN + kg + 1];
            const int krow = k4 * 4 + 2 * half;
            #pragma unroll
            for (int nt = 0; nt < 4; ++nt) {
                v2f bf;
                bf.x = Xs[krow * LDS_STRIDE + nt * 16 + l15];
                bf.y = Xs[(krow + 1) * LDS_STRIDE + nt * 16 + l15];
                acc[0][nt] = wmma4(a0, bf, acc[0][nt]);
                acc[1][nt] = wmma4(a1, bf, acc[1][nt]);
            }
        }
    }

    const float invs = rsqrtf(1.0f + 1e-5f);
    #pragma unroll
    for (int mi = 0; mi < 2; ++mi) {
        #pragma unroll
        for (int nt = 0; nt < 4; ++nt) {
            #pragma unroll
            for (int r = 0; r < 8; ++r) {
                const int o = m0 + mi * 16 + r + 8 * half;
                const int n = n0 + nt * 16 + l15;
                float v = acc[mi][nt][r] + b_sp[o];
                v = v * (gamma_sp[o] * invs) + beta_sp[o];
                out[((size_t)b * SP_OUT + o) * N_BEF + n] = gelu_tanh(v);
            }
        }
    }
}

// ---------------------------------------------------------------------------
// Kernel 4: dense branch. ConvTranspose1d(k=4,s=2,p=3) over the padded,
// gathered+interpolated sequence, as two phase GEMMs (even/odd outputs):
//   even[o,m] = sum_c w3[c,o]*x[c,m]   + w1[c,o]*x[c,m+1]
//   odd [o,m] = sum_c w2[c,o]*x[c,m+1] + w0[c,o]*x[c,m+2]
// x never materialized: Y[k2][j] = x[k2&127][m0+j+(k2>>7)] is gathered from
// dense_fea (L2-resident) while staging each LDS tile. Ntile=64 (+1 col so
// the odd phase reads a shifted window).
// ---------------------------------------------------------------------------
__global__ __launch_bounds__(256) void dense_convt_kernel(
    const float* __restrict__ dense_fea, const float* __restrict__ wcat,
    const float* __restrict__ b_ct, const float* __restrict__ gamma_dn,
    const float* __restrict__ beta_dn,
    const int* __restrict__ idx0, const int* __restrict__ idx1,
    const float* __restrict__ wA, const float* __restrict__ wB,
    float* __restrict__ out)
{
    __shared__ float Ys[32 * LDS_STRIDE];
    const int b = blockIdx.y;
    const int m0 = blockIdx.x * 64;
    const int tid = threadIdx.x;
    const int wv = tid >> 5, lane = tid & 31, half = lane >> 4, l15 = lane & 15;

    const float* dfb = dense_fea + (size_t)b * DN_CH * N_STK * N_PNT;
    const int*   I0 = idx0 + b * N_BEF;
    const int*   I1 = idx1 + b * N_BEF;
    const float* W0 = wA + b * N_BEF;
    const float* W1 = wB + b * N_BEF;

    v8f accE[4], accO[4];
    #pragma unroll
    for (int j = 0; j < 4; ++j) { v8f z = {}; accE[j] = z; accO[j] = z; }

    const int orow = wv * 16;
    for (int kk = 0; kk < 2 * DN_CH; kk += 32) {
        __syncthreads();
        for (int e = tid; e < 32 * 65; e += 256) {
            const int kr = e / 65, col = e % 65;
            const int k2 = kk + kr;
            const int c = k2 & 127, off = k2 >> 7;
            const int mp = m0 + col + off;          // padded-seq position [0,16385]
            int q = mp - 1;                         // edge pad: clamp to [0,16383]
            q = q < 0 ? 0 : (q > LSEQ - 1 ? LSEQ - 1 : q);
            const int s = q >> 5, p = q & 31;
            const float* fr = dfb + (size_t)c * N_STK * N_PNT;
            const float v = W0[s] * fr[I0[s] * N_PNT + p]
                          + W1[s] * fr[I1[s] * N_PNT + p];
            Ys[kr * LDS_STRIDE + col] = v;
        }
        __syncthreads();
        #pragma unroll
        for (int k4 = 0; k4 < 8; ++k4) {
            const int kg = kk + k4 * 4 + 2 * half;
            v2f aE, aO;
            aE.x = wcat[(size_t)(orow + l15) * 256 + kg];
            aE.y = wcat[(size_t)(orow + l15) * 256 + kg + 1];
            aO.x = wcat[(size_t)(DN_CH + orow + l15) * 256 + kg];
            aO.y = wcat[(size_t)(DN_CH + orow + l15) * 256 + kg + 1];
            const int krow = k4 * 4 + 2 * half;
            #pragma unroll
            for (int nt = 0; nt < 4; ++nt) {
                const int cb = nt * 16 + l15;
                v2f bE, bO;
                bE.x = Ys[krow * LDS_STRIDE + cb];
                bE.y = Ys[(krow + 1) * LDS_STRIDE + cb];
                bO.x = Ys[krow * LDS_STRIDE + cb + 1];
                bO.y = Ys[(krow + 1) * LDS_STRIDE + cb + 1];
                accE[nt] = wmma4(aE, bE, accE[nt]);
                accO[nt] = wmma4(aO, bO, accO[nt]);
            }
        }
    }

    const float invs = rsqrtf(1.0f + 1e-5f);
    #pragma unroll
    for (int nt = 0; nt < 4; ++nt) {
        #pragma unroll
        for (int r = 0; r < 8; ++r) {
            const int o = orow + r + 8 * half;
            const int m = m0 + nt * 16 + l15;
            const float sc = gamma_dn[o] * invs;
            const float be = beta_dn[o];
            const float bb = b_ct[o];
            float vE = (accE[nt][r] + bb) * sc + be;
            float vO = (accO[nt][r] + bb) * sc + be;
            float* orow_ptr = out + ((size_t)b * DN_CH + o) * (2 * LSEQ);
            orow_ptr[2 * m]     = gelu_tanh(vE);
            orow_ptr[2 * m + 1] = gelu_tanh(vO);
        }
    }
}

// ---------------------------------------------------------------------------
extern "C" void kernel_launch(void* const* d_in, const int* in_sizes, int n_in,
                              void* d_out, int out_size, void* d_ws, size_t ws_size,
                              hipStream_t stream) {
    const float* sparse_fea   = (const float*)d_in[0];
    const float* dense_fea    = (const float*)d_in[1];
    const float* stk_coor     = (const float*)d_in[2];
    const float* stk_coor_bef = (const float*)d_in[3];
    const float* w_sp     = (const float*)d_in[4];
    const float* b_sp     = (const float*)d_in[5];
    const float* gamma_sp = (const float*)d_in[6];
    const float* beta_sp  = (const float*)d_in[7];
    const float* w_ct     = (const float*)d_in[8];
    const float* b_ct     = (const float*)d_in[9];
    const float* gamma_dn = (const float*)d_in[10];
    const float* beta_dn  = (const float*)d_in[11];

    float* out = (float*)d_out;

    // workspace layout (393 KB): idx0,idx1 [8192 int], wA,wB [8192 f32], wcat [65536 f32]
    int*   idx0 = (int*)d_ws;
    int*   idx1 = idx0 + BS * N_BEF;
    float* wAp  = (float*)(idx1 + BS * N_BEF);
    float* wBp  = wAp + BS * N_BEF;
    float* wcat = wBp + BS * N_BEF;

    topk_kernel<<<(BS * N_BEF) / 8, 256, 0, stream>>>(stk_coor, stk_coor_bef,
                                                      idx0, idx1, wAp, wBp);
    wcat_kernel<<<256, 256, 0, stream>>>(w_ct, wcat);
    sparse_gemm_kernel<<<dim3(N_BEF / 64, BS), 256, 0, stream>>>(
        sparse_fea, w_sp, b_sp, gamma_sp, beta_sp, idx0, idx1, wAp, wBp, out);
    dense_convt_kernel<<<dim3(LSEQ / 64, BS), 256, 0, stream>>>(
        dense_fea, wcat, b_ct, gamma_dn, beta_dn, idx0, idx1, wAp, wBp,
        out + (size_t)BS * SP_OUT * N_BEF);
}